// RegionLoss_80023830659720
// MI455X (gfx1250) — compile-verified
//
#include <hip/hip_runtime.h>
#include <math.h>

// Problem constants (match reference)
#define Bn 64
#define Ln 50
#define Hh 26
#define Ww 26
#define NAn 5
#define NCn 80
#define CHn (NCn + 5)      // 85
#define HWn (Hh * Ww)      // 676
#define BLn (Bn * Ln)      // 3200
#define RECF 16            // floats per GT record (64 B)
#define NPART (Bn * NAn + BLn)   // 320 + 3200 = 3520 = 55 * 64 exactly

typedef __attribute__((ext_vector_type(2))) float v2f;
typedef __attribute__((ext_vector_type(8))) float v8f;

__constant__ float c_anc[NAn * 2] = {
    1.3221f, 1.73145f, 3.19275f, 4.00944f, 5.05587f,
    8.09892f, 9.47112f, 4.84053f, 11.2364f, 10.0071f};

__device__ __forceinline__ float sigmoidf_(float x) { return 1.0f / (1.0f + expf(-x)); }

// Darknet-style IoU on [cx,cy,w,h]
__device__ __forceinline__ float bbox_iou_(float x1, float y1, float w1, float h1,
                                           float x2, float y2, float w2, float h2) {
  float uw = fmaxf(x1 + 0.5f * w1, x2 + 0.5f * w2) - fminf(x1 - 0.5f * w1, x2 - 0.5f * w2);
  float uh = fmaxf(y1 + 0.5f * h1, y2 + 0.5f * h2) - fminf(y1 - 0.5f * h1, y2 - 0.5f * h2);
  float cw = w1 + w2 - uw;
  float ch = h1 + h2 - uh;
  float inter = (cw > 0.0f && ch > 0.0f) ? cw * ch : 0.0f;
  float uni = w1 * h1 + w2 * h2 - inter;
  return inter / fmaxf(uni, 1e-12f);
}

// ---------------- K1: per-GT record setup -------------------------------
// rec[t*16]: [0..3]=gt box (grid units), [4..7]=tx,ty,tw,th, [8]=tconf,
// [10]=cellcode int bits (best<<16 | gj<<8 | gi) or -1, [11]=cls int bits,
// [12]=valid flag
__global__ void k_gt_setup(const float* __restrict__ pred,
                           const float* __restrict__ target,
                           float* __restrict__ rec) {
  int t = blockIdx.x * blockDim.x + threadIdx.x;
  if (t >= BLn) return;
  int b = t / Ln;
  const float* tg = target + (size_t)t * 5;
  float cls = tg[0];
  float gx = tg[1] * Ww, gy = tg[2] * Hh, gw = tg[3] * Ww, gh = tg[4] * Hh;
  bool valid = tg[1] > 0.0f;

  // best anchor by wh-overlap ratio (first max wins, as jnp.argmax)
  int best = 0;
  float bestr = -1.0f;
  for (int a = 0; a < NAn; ++a) {
    float aw = c_anc[2 * a], ah = c_anc[2 * a + 1];
    float inter = fminf(gw, aw) * fminf(gh, ah);
    float uni = gw * gh + aw * ah - inter;
    float r = inter / fmaxf(uni, 1e-12f);
    if (r > bestr) { bestr = r; best = a; }
  }
  float fgi = floorf(gx), fgj = floorf(gy);
  float tx = gx - fgi, ty = gy - fgj;
  int gi = min(max((int)fgi, 0), Ww - 1);
  int gj = min(max((int)fgj, 0), Hh - 1);
  float aw = c_anc[2 * best], ah = c_anc[2 * best + 1];
  float tw = logf(fmaxf(gw, 1e-12f) / aw);
  float th = logf(fmaxf(gh, 1e-12f) / ah);

  // decoded pred box at (b, best, gj, gi)
  size_t base = ((size_t)(b * NAn + best) * CHn) * HWn + (size_t)gj * Ww + gi;
  float p0 = pred[base + 0 * HWn];
  float p1 = pred[base + 1 * HWn];
  float p2 = pred[base + 2 * HWn];
  float p3 = pred[base + 3 * HWn];
  float bx = sigmoidf_(p0) + (float)gi;
  float by = sigmoidf_(p1) + (float)gj;
  float bw = expf(p2) * aw;
  float bh = expf(p3) * ah;
  float tconf = bbox_iou_(gx, gy, gw, gh, bx, by, bw, bh);

  float* r = rec + (size_t)t * RECF;
  r[0] = gx; r[1] = gy; r[2] = gw; r[3] = gh;
  r[4] = tx; r[5] = ty; r[6] = tw; r[7] = th;
  r[8] = tconf;
  r[9] = 0.0f;
  r[10] = __int_as_float(valid ? ((best << 16) | (gj << 8) | gi) : -1);
  r[11] = __int_as_float((int)cls);
  r[12] = valid ? 1.0f : 0.0f;
  r[13] = 0.0f; r[14] = 0.0f; r[15] = 0.0f;
}

// ---------------- K2: per-cell coord + conf losses ----------------------
// One block per (b, a). Stages the batch's 50 GT records into LDS through
// the CDNA5 async global->LDS engine (ASYNCcnt), then fuses decode,
// 50-GT IoU max, GT-cell match, and the squared-error terms.
__global__ void k_main(const float* __restrict__ pred,
                       const float* __restrict__ rec,
                       float* __restrict__ blockpart) {
  int b = blockIdx.x / NAn;
  int a = blockIdx.x % NAn;
  int tid = threadIdx.x;

  __shared__ alignas(16) float rs[Ln * RECF];  // 3200 B of GT records
  {
    // 200 x 16B chunks; one async b128 per active lane.
    int chunk = tid;
    if (chunk < (Ln * RECF) / 4) {
      unsigned lds = (unsigned)(size_t)(&rs[chunk * 4]);
      const float* g = rec + (size_t)b * Ln * RECF + (size_t)chunk * 4;
      asm volatile("global_load_async_to_lds_b128 %0, %1, off"
                   :: "v"(lds), "v"(g) : "memory");
    }
    asm volatile("s_wait_asynccnt 0" ::: "memory");
  }
  __syncthreads();

  float aw = c_anc[2 * a], ah = c_anc[2 * a + 1];
  float lsum = 0.0f;

  for (int cell = tid; cell < HWn; cell += blockDim.x) {
    int j = cell / Ww, i = cell - j * Ww;
    size_t base = ((size_t)(b * NAn + a) * CHn) * HWn + cell;
    float p0 = pred[base];
    float p1 = pred[base + HWn];
    float p2 = pred[base + 2 * HWn];
    float p3 = pred[base + 3 * HWn];
    float p4 = pred[base + 4 * HWn];
    float sx = sigmoidf_(p0), sy = sigmoidf_(p1), pc = sigmoidf_(p4);
    float bx = sx + (float)i, by = sy + (float)j;
    float bw = expf(p2) * aw, bh = expf(p3) * ah;

    int code = (a << 16) | (j << 8) | i;
    float maxiou = 0.0f;
    float tbx = 0.5f, tby = 0.5f, tbw = 0.0f, tbh = 0.0f, tcf = 0.0f; // EPOCH==0 defaults
    bool isgt = false;
    for (int l = 0; l < Ln; ++l) {
      const float* r = &rs[l * RECF];
      if (r[12] != 0.0f) {
        float iou = bbox_iou_(r[0], r[1], r[2], r[3], bx, by, bw, bh);
        maxiou = fmaxf(maxiou, iou);
        if (__float_as_int(r[10]) == code) {
          isgt = true;
          tbx = r[4]; tby = r[5]; tbw = r[6]; tbh = r[7]; tcf = r[8];
        }
      }
    }
    float confm = (maxiou > 0.6f) ? 0.0f : 1.0f;  // NO_OBJECT_SCALE
    if (isgt) confm = 5.0f;                        // OBJECT_SCALE
    float dx = sx - tbx, dy = sy - tby, dw = p2 - tbw, dh = p3 - tbh;
    float dc = pc - tcf;
    lsum += 0.5f * (dx * dx + dy * dy + dw * dw + dh * dh)
          + 0.5f * confm * dc * dc;
  }

  // deterministic block reduction
  for (int o = 16; o > 0; o >>= 1) lsum += __shfl_xor(lsum, o, 32);
  __shared__ float wsum[8];
  int wid = tid >> 5;
  if ((tid & 31) == 0) wsum[wid] = lsum;
  __syncthreads();
  if (tid == 0) {
    float s = 0.0f;
    int nw = blockDim.x >> 5;
    for (int k = 0; k < nw; ++k) s += wsum[k];
    blockpart[blockIdx.x] = s;
  }
}

// ---------------- K3: class CE at GT cells (one wave32 per GT) ----------
__global__ void k_cls(const float* __restrict__ pred,
                      const float* __restrict__ rec,
                      float* __restrict__ cepart) {
  int wave = (blockIdx.x * blockDim.x + threadIdx.x) >> 5;
  int lane = threadIdx.x & 31;
  if (wave >= BLn) return;
  const float* r = rec + (size_t)wave * RECF;
  int code = __float_as_int(r[10]);
  float valid = r[12];
  float ce = 0.0f;
  if (valid != 0.0f) {
    int b = wave / Ln;
    int a = (code >> 16) & 0xff, j = (code >> 8) & 0xff, i = code & 0xff;
    int cls = __float_as_int(r[11]);
    size_t base = ((size_t)(b * NAn + a) * CHn + 5) * HWn + (size_t)j * Ww + i;
    float v[3];
    int cnt = 0;
    float m = -3.4e38f;
    float pcls = 0.0f;
    for (int c = lane; c < NCn; c += 32) {
      float p = pred[base + (size_t)c * HWn];
      v[cnt++] = p;
      m = fmaxf(m, p);
      if (c == cls) pcls = p;
    }
    for (int o = 16; o > 0; o >>= 1) m = fmaxf(m, __shfl_xor(m, o, 32));
    float s = 0.0f;
    for (int k = 0; k < cnt; ++k) s += expf(v[k] - m);
    for (int o = 16; o > 0; o >>= 1) s += __shfl_xor(s, o, 32);
    for (int o = 16; o > 0; o >>= 1) pcls += __shfl_xor(pcls, o, 32);
    ce = logf(s) + m - pcls;  // CLASS_SCALE == 1
  }
  if (lane == 0) cepart[wave] = ce;
}

// ---------------- K4: deterministic WMMA final sum ----------------------
// Single wave32, EXEC all-ones, uniform flow (WMMA requirements).
// A = ones(16x4); each V_WMMA_F32_16X16X4_F32 accumulates the K-column sums
// of a 4x16 chunk of B into every row of D, so after 55 accumulating WMMAs
// row 0 (VGPR 0, lanes 0..15) holds 16 column partials whose fixed-order
// shuffle sum is the exact total. A=ones makes the result independent of
// the A/B lane->element mapping: every loaded value contributes exactly once.
__global__ void k_reduce_wmma(const float* __restrict__ vals, float* __restrict__ out) {
  int lane = threadIdx.x;  // blockDim == 32
  v2f a;
  a.x = 1.0f;
  a.y = 1.0f;
  v8f c = {};
  for (int chunk = 0; chunk < NPART / 64; ++chunk) {
    v2f bv;
    bv.x = vals[chunk * 64 + lane];
    bv.y = vals[chunk * 64 + 32 + lane];
    c = __builtin_amdgcn_wmma_f32_16x16x4_f32(
        /*neg_a=*/false, a, /*neg_b=*/false, bv,
        /*c_mod=*/(short)0, c, /*reuse_a=*/false, /*reuse_b=*/false);
  }
  float t = (lane < 16) ? c[0] : 0.0f;
  for (int o = 16; o > 0; o >>= 1) t += __shfl_xor(t, o, 32);
  if (lane == 0) out[0] = t;
}

extern "C" void kernel_launch(void* const* d_in, const int* in_sizes, int n_in,
                              void* d_out, int out_size, void* d_ws, size_t ws_size,
                              hipStream_t stream) {
  (void)in_sizes; (void)n_in; (void)out_size; (void)ws_size;
  const float* pred = (const float*)d_in[0];
  const float* target = (const float*)d_in[1];
  float* out = (float*)d_out;
  float* ws = (float*)d_ws;

  float* rec = ws;                        // 3200 * 16 floats = 204800 B
  float* parts = ws + (size_t)BLn * RECF; // 320 block partials + 3200 CE partials

  k_gt_setup<<<dim3((BLn + 255) / 256), dim3(256), 0, stream>>>(pred, target, rec);
  k_main<<<dim3(Bn * NAn), dim3(256), 0, stream>>>(pred, rec, parts);
  k_cls<<<dim3((BLn + 7) / 8), dim3(256), 0, stream>>>(pred, rec, parts + Bn * NAn);
  k_reduce_wmma<<<dim3(1), dim3(32), 0, stream>>>(parts, out);
}